// PARAFAC_39986145526078
// MI455X (gfx1250) — compile-verified
//
#include <hip/hip_runtime.h>

// PARAFAC / Khatri-Rao expansion: out[b,j,k] = sum_r U[r,idx_b] * V[r,j] * W[r,k]
// Per batch b this is a (J x R) x (R x K) GEMM with A[j,r] = V[r,j]*U[r,idx_b].
// Store-bandwidth bound: 268 MB f32 output vs 4.29 GFLOP (AI ~16 FLOP/B).
// Uses V_WMMA_F32_16X16X4_F32 (full fp32 precision) + non-temporal stores.

typedef __attribute__((ext_vector_type(2))) float v2f;
typedef __attribute__((ext_vector_type(8))) float v8f;

#define RANK  32
#define I_DIM 1024
#define J_DIM 512
#define K_DIM 512
#define B_DIM 256

__global__ __launch_bounds__(256) void parafac_wmma_f32(
    const float* __restrict__ U,     // (RANK, I)
    const float* __restrict__ V,     // (RANK, J)
    const float* __restrict__ W,     // (RANK, K)
    const int*   __restrict__ index, // (B,)
    float*       __restrict__ out)   // (B, J, K)
{
    const int lane = threadIdx.x & 31;
    const int wave = threadIdx.x >> 5;   // 0..7
    const int j0   = blockIdx.x * 16;    // j-strip for this workgroup
    const int b    = blockIdx.y;
    const int idx  = index[b];           // uniform per block

    const int half = lane >> 4;          // 0: lanes 0-15, 1: lanes 16-31
    const int l15  = lane & 15;
    const int jj   = j0 + l15;           // A-matrix row M for this lane
    const int rb   = half * 2;           // K-offset within 4-chunk: 0 or 2

    // ---- Build A fragments: A[j, r] = V[r, j] * U[r, idx] ----
    // 32-bit 16x4 A layout (ISA 7.12.2): VGPR0 holds K = rb, VGPR1 holds K = rb+1.
    // 8 chunks cover the full R = 32 reduction.
    float a0[8], a1[8];
#pragma unroll
    for (int c = 0; c < 8; ++c) {
        const int r  = 4 * c + rb;
        const float u0 = U[r * I_DIM + idx];         // uniform -> scalar load
        const float u1 = U[(r + 1) * I_DIM + idx];
        a0[c] = V[r * J_DIM + jj] * u0;
        a1[c] = V[(r + 1) * J_DIM + jj] * u1;
    }

    // ---- Each wave computes 4 k-tiles: kt = wave + 8*t ----
#pragma unroll
    for (int t = 0; t < 4; ++t) {
        const int k0 = (wave + 8 * t) * 16;
        const int nn = k0 + l15;         // B/D-matrix column N for this lane

        v8f acc = {};
#pragma unroll
        for (int c = 0; c < 8; ++c) {
            const int r = 4 * c + rb;
            v2f A, Bv;
            A.x  = a0[c];
            A.y  = a1[c];
            Bv.x = W[r * K_DIM + nn];         // B 4x16, mirror of A layout
            Bv.y = W[(r + 1) * K_DIM + nn];
            // 8 args: (neg_a, A, neg_b, B, c_mod, C, reuse_a, reuse_b)
            acc = __builtin_amdgcn_wmma_f32_16x16x4_f32(
                false, A, false, Bv, (short)0, acc, false, false);
        }

        // ---- Store 16x16 f32 D tile ----
        // C/D layout: VGPR p -> row M = p (lanes 0-15) or M = p + 8 (lanes 16-31),
        // column N = lane & 15. Streamed output: non-temporal to spare L2.
        const size_t base = (size_t)b * (size_t)(J_DIM * K_DIM)
                          + (size_t)(j0 + half * 8) * K_DIM
                          + (size_t)nn;
#pragma unroll
        for (int p = 0; p < 8; ++p) {
            __builtin_nontemporal_store(acc[p], &out[base + (size_t)p * K_DIM]);
        }
    }
}

extern "C" void kernel_launch(void* const* d_in, const int* in_sizes, int n_in,
                              void* d_out, int out_size, void* d_ws, size_t ws_size,
                              hipStream_t stream) {
    (void)in_sizes; (void)n_in; (void)out_size; (void)d_ws; (void)ws_size;
    const float* U     = (const float*)d_in[0];
    const float* V     = (const float*)d_in[1];
    const float* W     = (const float*)d_in[2];
    const int*   index = (const int*)d_in[3];
    float*       out   = (float*)d_out;

    dim3 grid(J_DIM / 16, B_DIM, 1);   // 32 x 256 workgroups
    dim3 block(256, 1, 1);             // 8 wave32 waves
    parafac_wmma_f32<<<grid, block, 0, stream>>>(U, V, W, index, out);
}